// RoIPool_21749714387549
// MI455X (gfx1250) — compile-verified
//
#include <hip/hip_runtime.h>
#include <stdint.h>

// Problem constants (from reference): B=8, C=128, H=W=64, 16 cells/img, N=128.
#define BB        8
#define CC        128
#define HWPX      4096                 // 64*64
#define NCELLS    128
#define SLICES    4                    // blocks per cell
#define CH_PER_BLOCK (CC / SLICES)     // 32 channels per block
#define THREADS   256
#define WAVES     (THREADS / 32)       // 8 waves (wave32)
#define CH_PER_WAVE (CH_PER_BLOCK / WAVES) // 4 channels per wave

__global__ __launch_bounds__(THREADS)
void roipool_max_kernel(const float* __restrict__ fm,     // (B,C,H,W) f32
                        const int*   __restrict__ masks,  // (N,H,W) int32 0/1
                        const int*   __restrict__ counts, // (B,) int32
                        float*       __restrict__ out)    // (N,C) f32
{
    __shared__ int smask[HWPX];   // 16 KB: this cell's mask, staged via async DMA
    __shared__ int s_batch;

    const int tid   = (int)threadIdx.x;
    const int lane  = tid & 31;
    const int wave  = tid >> 5;
    const int n     = (int)blockIdx.x >> 2;          // cell id
    const int slice = (int)blockIdx.x & (SLICES - 1);

    // ---- batch id for this cell from cell_counts (prefix walk, jnp.repeat) ----
    if (tid == 0) {
        int acc = 0, b = BB - 1;
        for (int i = 0; i < BB; ++i) {
            int c = counts[i];
            if (n < acc + c) { b = i; break; }
            acc += c;
        }
        s_batch = b;
    }

    // ---- stage 16 KB of int32 mask into LDS with async global->LDS copies ----
    // GVS mode: mem = SADDR(64b sgpr) + VADDR(32b byte offset); LDS dst = VGPR.
    const int* mbase = masks + (size_t)n * HWPX;
    const uint32_t lds0 = (uint32_t)(uintptr_t)(void*)&smask[0]; // low 32b = LDS offset
    #pragma unroll
    for (int chunk = 0; chunk < 4; ++chunk) {
        uint32_t off  = (uint32_t)(chunk * (THREADS * 16) + tid * 16); // 4 KB/chunk
        uint32_t ldsa = lds0 + off;
        asm volatile("global_load_async_to_lds_b128 %0, %1, %2"
                     :
                     : "v"(ldsa), "v"(off), "s"(mbase)
                     : "memory");
    }
    asm volatile("s_wait_asynccnt 0x0" ::: "memory"); // ASYNCcnt -> 0
    __syncthreads();

    // ---- pack this lane's 128 mask bits (pixel q = (i*32+lane)*4 + e) ----
    uint32_t mb[4] = {0u, 0u, 0u, 0u};
    #pragma unroll
    for (int i = 0; i < 32; ++i) {
        const int4 m = ((const int4*)smask)[i * 32 + lane];
        uint32_t bits = (m.x != 0 ? 1u : 0u) | (m.y != 0 ? 2u : 0u) |
                        (m.z != 0 ? 4u : 0u) | (m.w != 0 ? 8u : 0u);
        mb[i >> 3] |= bits << ((i & 7) * 4);
    }

    const int b = s_batch;
    const float4* fmb = (const float4*)(fm + (size_t)b * CC * HWPX);
    const float NEG = -__builtin_inff();

    #pragma unroll
    for (int k = 0; k < CH_PER_WAVE; ++k) {
        const int c = slice * CH_PER_BLOCK + wave * CH_PER_WAVE + k;
        const float4* plane = fmb + (size_t)c * (HWPX / 4);

        if (k + 1 < CH_PER_WAVE) { // warm next channel plane (global_prefetch_b8)
            const char* np = (const char*)(plane + (HWPX / 4)) + lane * 512;
            __builtin_prefetch(np, 0, 0);
        }

        float mx = NEG;
        #pragma unroll 8
        for (int i = 0; i < 32; ++i) {
            float4 v = plane[i * 32 + lane];            // coalesced b128 stream
            uint32_t bits = (mb[i >> 3] >> ((i & 7) * 4)) & 0xFu;
            mx = fmaxf(mx, (bits & 1u) ? v.x : NEG);
            mx = fmaxf(mx, (bits & 2u) ? v.y : NEG);
            mx = fmaxf(mx, (bits & 4u) ? v.z : NEG);
            mx = fmaxf(mx, (bits & 8u) ? v.w : NEG);
        }

        // wave32 max-reduction (warpSize == 32 on gfx1250)
        #pragma unroll
        for (int o = 16; o > 0; o >>= 1)
            mx = fmaxf(mx, __shfl_xor(mx, o, 32));

        if (lane == 0)
            out[n * CC + c] = mx;
    }
}

extern "C" void kernel_launch(void* const* d_in, const int* in_sizes, int n_in,
                              void* d_out, int out_size, void* d_ws, size_t ws_size,
                              hipStream_t stream) {
    (void)in_sizes; (void)n_in; (void)out_size; (void)d_ws; (void)ws_size;
    const float* fm     = (const float*)d_in[0];   // feature_maps (B,C,H,W) f32
    const int*   masks  = (const int*)d_in[1];     // cell_masks (N,H,W) int
    const int*   counts = (const int*)d_in[2];     // cell_counts (B,) int32
    float*       out    = (float*)d_out;           // (N,C) f32

    roipool_max_kernel<<<dim3(NCELLS * SLICES), dim3(THREADS), 0, stream>>>(
        fm, masks, counts, out);
}